// EnergyOptimalNetwork_31361851195996
// MI455X (gfx1250) — compile-verified
//
#include <hip/hip_runtime.h>
#include <math.h>

typedef __attribute__((ext_vector_type(2))) float v2f;
typedef __attribute__((ext_vector_type(4))) float v4f;
typedef __attribute__((ext_vector_type(8))) float v8f;

#define B_  16
#define N_  65536
#define T_  64
#define K_  64
#define D_  32

__device__ __forceinline__ float xorf16(float v)       { return __shfl_xor(v, 16, 32); }
__device__ __forceinline__ unsigned xoru16(unsigned v) { return __shfl_xor(v, 16, 32); }

// Insert v into sorted-descending {m0..m4} with v_med3_f32: m_i' = med3(m_{i-1}, m_i, v)
__device__ __forceinline__ void ins5(float& m0, float& m1, float& m2, float& m3, float& m4, float v) {
    float n4 = __builtin_amdgcn_fmed3f(m3, m4, v);
    float n3 = __builtin_amdgcn_fmed3f(m2, m3, v);
    float n2 = __builtin_amdgcn_fmed3f(m1, m2, v);
    float n1 = __builtin_amdgcn_fmed3f(m0, m1, v);
    m0 = fmaxf(m0, v);
    m1 = n1; m2 = n2; m3 = n3; m4 = n4;
}

__global__ __launch_bounds__(128) void eon_kernel(
    const float* __restrict__ hist,
    const float* __restrict__ feat,
    const float* __restrict__ dict,
    const float* __restrict__ thr,
    const float* __restrict__ memb,
    const float* __restrict__ refrac,
    const float* __restrict__ energy,
    const float* __restrict__ metab,
    float* __restrict__ out)
{
    const int lane = threadIdx.x & 31;
    const int wave = threadIdx.x >> 5;
    const int n    = blockIdx.x * 4 + wave;
    const int half = lane >> 4;   // K-pair of the 16x16x4 fragment / tail work split
    const int row  = lane & 15;   // M row (A) / b column (B,D)

    // Prefetch the hist tail we need much later (hidden behind the matmul).
    __builtin_prefetch(hist + ((size_t)row * N_ + n) * T_ + 54, 0, 0);

    // ---- B fragments: feat^T (32 x 16), 8 k-steps of 4. float2 per lane. ----
    const float* fbase = feat + ((size_t)row * N_ + n) * D_ + half * 2;
    v2f bfrag[8];
    #pragma unroll
    for (int s = 0; s < 8; ++s)
        bfrag[s] = *(const v2f*)(fbase + s * 4);

    // ---- 4 accumulator tiles: projT[k, b], k-tile t covers rows t*16..t*16+15 ----
    v8f acc[4];
    #pragma unroll
    for (int t = 0; t < 4; ++t)
        #pragma unroll
        for (int r = 0; r < 8; ++r) acc[t][r] = 0.0f;

    const float* abase = dict + (size_t)n * K_ * D_ + half * 2;
    #pragma unroll
    for (int t = 0; t < 4; ++t) {
        const float* ab = abase + (size_t)(t * 16 + row) * D_;
        #pragma unroll
        for (int s = 0; s < 8; ++s) {
            v2f a = *(const v2f*)(ab + s * 4);
            acc[t] = __builtin_amdgcn_wmma_f32_16x16x4_f32(
                false, a, false, bfrag[s], (short)0, acc[t], false, false);
        }
    }
    // acc[t][r] = proj[b = row][k = t*16 + r + half*8]; lane pair (l, l^16) holds all 64 k.

    // ---- top-5 of |proj| over the 64 k values (exact 0.95-quantile of 64) ----
    float m0 = -1.f, m1 = -1.f, m2 = -1.f, m3 = -1.f, m4 = -1.f;
    #pragma unroll
    for (int t = 0; t < 4; ++t)
        #pragma unroll
        for (int r = 0; r < 8; ++r)
            ins5(m0, m1, m2, m3, m4, fabsf(acc[t][r]));

    {   // merge partner lane's sorted top-5
        float o0 = xorf16(m0), o1 = xorf16(m1), o2 = xorf16(m2), o3 = xorf16(m3), o4 = xorf16(m4);
        ins5(m0, m1, m2, m3, m4, o0);
        ins5(m0, m1, m2, m3, m4, o1);
        ins5(m0, m1, m2, m3, m4, o2);
        ins5(m0, m1, m2, m3, m4, o3);
        ins5(m0, m1, m2, m3, m4, o4);
    }
    // sorted ascending v[0..63]: q = v[59] + frac*(v[60]-v[59]); v[60]=m3, v[59]=m4
    const float frac = 0.95f * 63.0f - 59.0f;
    const float q  = m4 + frac * (m3 - m4);
    const float q1 = q + 1e-6f;

    // ---- soft-threshold stats, closed form from the top-5 ----
    // Ranks <= 59 satisfy ap <= v[59] <= q: they contribute nothing to either stat.
    float active = 0.0f, sumasp = 0.0f;
    {
        float mm[4] = { m0, m1, m2, m3 };
        #pragma unroll
        for (int i = 0; i < 4; ++i) {
            sumasp += fmaxf(mm[i] - q, 0.0f);
            active += (mm[i] > q1) ? 1.0f : 0.0f;
        }
    }
    const float sparse_info = active * (sumasp * (1.0f / 64.0f));

    // ================= per-(b,n) tail, split across the lane pair =================
    const int b = row;
    const size_t hb = ((size_t)b * N_ + n) * T_;
    float r[10];
    {
        v2f h0 = *(const v2f*)(hist + hb + 54);
        v4f h1 = *(const v4f*)(hist + hb + 56);
        v4f h2 = *(const v4f*)(hist + hb + 60);
        r[0] = h0.x; r[1] = h0.y;
        r[2] = h1.x; r[3] = h1.y; r[4] = h1.z; r[5] = h1.w;
        r[6] = h2.x; r[7] = h2.y; r[8] = h2.z; r[9] = h2.w;
    }

    const float th   = thr[n];
    const float last = r[9];
    const float slope = (r[9] - r[0]) * (1.0f / 9.0f);   // telescoped mean of diffs
    const float predicted = last + slope;
    const float pei = fabsf(slope) / (fabsf(predicted) + 1e-6f); // |last-predicted|=|slope|
    const float sg  = 1.0f / (1.0f + __expf(-(last - th)));
    const float surprise = -__logf(sg + 1e-8f);

    // ---- 2x10 contingency table: 4-bit packed counters, 5 timesteps per half-lane ----
    float xx[5];
    #pragma unroll
    for (int k = 0; k < 5; ++k) xx[k] = half ? r[5 + k] : r[k];

    unsigned c0 = 0, c1 = 0, c2 = 0;   // cats 0-7 / 8-15 / 16-19, 4-bit fields
    #pragma unroll
    for (int k = 0; k < 5; ++k) {
        float x = xx[k];
        // bin = digitize(x, linspace(-3,3,10)) = clamp(floor(1.5x+4.5)+1, 0, 10)
        float bf = __builtin_amdgcn_fmed3f(0.0f, floorf(fmaf(x, 1.5f, 4.5f)) + 1.0f, 10.0f);
        int bin = (int)bf;
        int col = (bin == 0) ? 9 : (bin - 1);
        int cat = col + ((x > th) ? 10 : 0);
        unsigned inc = (bin < 10) ? (1u << ((cat & 7) * 4)) : 0u;   // weight = valid
        int hi3 = cat >> 3;
        c0 += (hi3 == 0) ? inc : 0u;
        c1 += (hi3 == 1) ? inc : 0u;
        c2 += (hi3 == 2) ? inc : 0u;
    }
    // merge the two half-lane histograms (fields <= 10, no carry)
    c0 += xoru16(c0);
    c1 += xoru16(c1);
    c2 += xoru16(c2);

    float cnt[20];
    #pragma unroll
    for (int c = 0; c < 20; ++c) {
        unsigned reg = (c < 8) ? c0 : ((c < 16) ? c1 : c2);
        cnt[c] = (float)((reg >> ((c & 7) * 4)) & 0xFu) + 1e-6f;
    }
    float ps0 = 0.0f, ps1 = 0.0f;
    #pragma unroll
    for (int c = 0; c < 10; ++c) { ps0 += cnt[c]; ps1 += cnt[c + 10]; }
    const float inv = 1.0f / (ps0 + ps1);

    // each half-lane computes its spike-row's 10 log terms
    const float psh = (half ? ps1 : ps0) * inv;
    float mi_p = 0.0f;
    #pragma unroll
    for (int c = 0; c < 10; ++c) {
        float pc  = (half ? cnt[c + 10] : cnt[c]) * inv;
        float pic = (cnt[c] + cnt[c + 10]) * inv;
        mi_p += pc * __logf(pc / (psh * pic + 1e-10f));
    }
    const float mi = fmaxf(mi_p + xorf16(mi_p), 0.0f);

    const float info = fmaxf(0.4f * surprise + 0.3f * mi + 0.2f * pei + 0.1f * sparse_info, 0.0f);

    const size_t idx = (size_t)b * N_ + n;
    const float rc = refrac[idx];
    const float md = metab[idx];
    const float mp = memb[idx];
    const float ec = energy[idx];

    const float imm     = (1e-12f + ((rc > 0.0f) ? 5e-16f * rc : 0.0f)) * md;
    const float delayed = 1e-15f + 0.7f * 1e-12f * 1.1f + 0.1f * 1e-15f * fabsf(mp);
    const float imm_eff = info / imm;
    const float del_eff = info * 0.8f / delayed;
    float timing = (imm_eff >= del_eff) ? 0.5f : 1.5f;
    timing *= (info > 2.0f) ? 0.8f : ((info < 0.5f) ? 1.5f : 1.0f);
    const bool feasible = (ec + imm) <= 1e-6f;
    const bool should   = (timing < 1.0f) && feasible && (rc <= 0.0f);

    if (half == 0) {
        out[idx]                       = should ? 1.0f : 0.0f;
        out[(size_t)B_ * N_ + idx]     = should ? timing : INFINITY;
        out[2 * (size_t)B_ * N_ + idx] = info;
    }
}

extern "C" void kernel_launch(void* const* d_in, const int* in_sizes, int n_in,
                              void* d_out, int out_size, void* d_ws, size_t ws_size,
                              hipStream_t stream) {
    (void)in_sizes; (void)n_in; (void)out_size; (void)d_ws; (void)ws_size;
    const float* hist   = (const float*)d_in[0];
    const float* feat   = (const float*)d_in[1];
    const float* dict   = (const float*)d_in[2];
    const float* thr    = (const float*)d_in[3];
    const float* memb   = (const float*)d_in[4];
    const float* refrac = (const float*)d_in[5];
    const float* energy = (const float*)d_in[6];
    const float* metab  = (const float*)d_in[7];
    float* out = (float*)d_out;

    dim3 grid(N_ / 4);   // one wave32 per n, 4 waves per block
    dim3 block(128);
    eon_kernel<<<grid, block, 0, stream>>>(hist, feat, dict, thr, memb,
                                           refrac, energy, metab, out);
}